// GCL_12592844112145
// MI455X (gfx1250) — compile-verified
//
#include <hip/hip_runtime.h>

typedef __attribute__((ext_vector_type(2))) float v2f;
typedef __attribute__((ext_vector_type(4))) float v4f;
typedef __attribute__((ext_vector_type(8))) float v8f;

#define IN_NF  64
#define HID_NF 64
#define KDIM   128   // 2*IN_NF = HID_NF+IN_NF

#define WMMA_F32(a, b, c) \
  __builtin_amdgcn_wmma_f32_16x16x4_f32(false, (a), false, (b), (short)0, (c), false, false)

// ---------------- zero the aggregation workspace ----------------
__global__ void gnn_zero_kernel(float* __restrict__ p, int n4) {
  v4f z = {};
  for (int i = blockIdx.x * blockDim.x + threadIdx.x; i < n4;
       i += gridDim.x * blockDim.x)
    ((v4f*)p)[i] = z;
}

// Stage W[128][64] into LDS as interleaved K-pairs with bank swizzle:
//   sW2[p*64 + (col ^ ((p&1)<<4))] = { W[2p][col], W[2p+1][col] }
// One B fragment == one conflict-free ds_load_b64 into an even VGPR pair.
__device__ __forceinline__ void stage_weights(v2f* sW2, const float* __restrict__ W) {
  const int t = threadIdx.x;
#pragma unroll
  for (int i = 0; i < 8; ++i) {
    const int q2  = t + i * 256;        // 0..2047 : (row-pair, even col) items
    const int p   = q2 >> 5;            // row pair 0..63
    const int col = (q2 & 31) * 2;      // even logical column
    v2f a0 = *(const v2f*)(W + (size_t)(2 * p)     * HID_NF + col);
    v2f a1 = *(const v2f*)(W + (size_t)(2 * p + 1) * HID_NF + col);
    const int colPhys = col ^ ((p & 1) << 4);
    v4f pk; pk.x = a0.x; pk.y = a1.x; pk.z = a0.y; pk.w = a1.y;
    *(v4f*)(sW2 + p * 64 + colPhys) = pk;
  }
}

// ---------------- edge model + fused scatter-add ----------------
// One wave32 per 16-edge tile: D = A(16x128 gathered) x B(128x64 LDS),
// 32 K-steps of V_WMMA_F32_16X16X4_F32 over 4 N-tiles, then relu+atomic scatter.
__global__ void __launch_bounds__(256)
gnn_edge_kernel(const float* __restrict__ h,       // [N, 64]
                const int*   __restrict__ eidx,    // [2, E] (row; col)
                const float* __restrict__ W,       // [128, 64]
                const float* __restrict__ bias,    // [64]
                float*       __restrict__ agg,     // [N, 64]
                int E)
{
  __shared__ v2f sW2[64 * 64];                     // 32 KB
  stage_weights(sW2, W);
  __syncthreads();

  const int lane = threadIdx.x & 31;
  const int wave = threadIdx.x >> 5;
  const int te   = (blockIdx.x * 8 + wave) * 16;   // edge tile base
  const int m    = lane & 15;                      // A row / B,C column-in-tile
  const int half = lane >> 4;                      // 0 -> K+0/1, 1 -> K+2/3
  const int koff = half * 2;

  const int e  = te + m;
  const int ec = (e < E) ? e : (E - 1);            // clamp: keep EXEC all-ones
  const int r  = eidx[ec];                         // row (scatter target)
  const int c  = eidx[E + ec];                     // col
  const float* pR = h + (size_t)r * IN_NF;
  const float* pC = h + (size_t)c * IN_NF;

  // B fragment base pointers (swizzle folded in): even/odd N-tiles
  const v2f* pE = sW2 + half * 64 + half * 16 + m;        // n = 0, 2
  const v2f* pO = sW2 + half * 64 + (16 - half * 16) + m; // n = 1, 3

  v8f acc[4] = {};

  for (int k = 0; k < 16; ++k) {                   // K = 0..63 : h[row]
    v2f a  = *(const v2f*)(pR + 4 * k + koff);
    v2f b0 = pE[k * 128];
    v2f b1 = pO[k * 128];
    v2f b2 = pE[k * 128 + 32];
    v2f b3 = pO[k * 128 + 32];
    acc[0] = WMMA_F32(a, b0, acc[0]);
    acc[1] = WMMA_F32(a, b1, acc[1]);
    acc[2] = WMMA_F32(a, b2, acc[2]);
    acc[3] = WMMA_F32(a, b3, acc[3]);
  }
  for (int k = 16; k < 32; ++k) {                  // K = 64..127 : h[col]
    v2f a  = *(const v2f*)(pC + 4 * (k - 16) + koff);
    v2f b0 = pE[k * 128];
    v2f b1 = pO[k * 128];
    v2f b2 = pE[k * 128 + 32];
    v2f b3 = pO[k * 128 + 32];
    acc[0] = WMMA_F32(a, b0, acc[0]);
    acc[1] = WMMA_F32(a, b1, acc[1]);
    acc[2] = WMMA_F32(a, b2, acc[2]);
    acc[3] = WMMA_F32(a, b3, acc[3]);
  }

  // bias + ReLU + scatter-add (hardware f32 atomics into L2)
#pragma unroll
  for (int n = 0; n < 4; ++n) {
    const int   colI = n * 16 + m;
    const float bv   = bias[colI];
#pragma unroll
    for (int rr = 0; rr < 8; ++rr) {
      const int eoff = rr + half * 8;              // edge-in-tile of this C elem
      float v = acc[n][rr] + bv;
      v = v > 0.0f ? v : 0.0f;
      const int tgt = __shfl(r, eoff, 32);         // row index of that edge
      if (te + eoff < E)
        unsafeAtomicAdd(agg + (size_t)tgt * HID_NF + colI, v);
    }
  }
}

// ---------------- node model ----------------
__global__ void __launch_bounds__(256)
gnn_node_kernel(const float* __restrict__ h,       // [N, 64]
                const float* __restrict__ agg,     // [N, 64]
                const float* __restrict__ W,       // [128, 64]
                const float* __restrict__ bias,    // [64]
                float*       __restrict__ out,     // [N, 64]
                int N)
{
  __shared__ v2f sW2[64 * 64];
  stage_weights(sW2, W);
  __syncthreads();

  const int lane = threadIdx.x & 31;
  const int wave = threadIdx.x >> 5;
  const int tn   = (blockIdx.x * 8 + wave) * 16;   // node tile base
  const int m    = lane & 15;
  const int half = lane >> 4;
  const int koff = half * 2;

  const int nd  = tn + m;
  const int ndc = (nd < N) ? nd : (N - 1);
  const float* pR = h   + (size_t)ndc * IN_NF;
  const float* pC = agg + (size_t)ndc * HID_NF;

  const v2f* pE = sW2 + half * 64 + half * 16 + m;
  const v2f* pO = sW2 + half * 64 + (16 - half * 16) + m;

  v8f acc[4] = {};

  for (int k = 0; k < 16; ++k) {                   // K = 0..63 : h
    v2f a  = *(const v2f*)(pR + 4 * k + koff);
    v2f b0 = pE[k * 128];
    v2f b1 = pO[k * 128];
    v2f b2 = pE[k * 128 + 32];
    v2f b3 = pO[k * 128 + 32];
    acc[0] = WMMA_F32(a, b0, acc[0]);
    acc[1] = WMMA_F32(a, b1, acc[1]);
    acc[2] = WMMA_F32(a, b2, acc[2]);
    acc[3] = WMMA_F32(a, b3, acc[3]);
  }
  for (int k = 16; k < 32; ++k) {                  // K = 64..127 : agg
    v2f a  = *(const v2f*)(pC + 4 * (k - 16) + koff);
    v2f b0 = pE[k * 128];
    v2f b1 = pO[k * 128];
    v2f b2 = pE[k * 128 + 32];
    v2f b3 = pO[k * 128 + 32];
    acc[0] = WMMA_F32(a, b0, acc[0]);
    acc[1] = WMMA_F32(a, b1, acc[1]);
    acc[2] = WMMA_F32(a, b2, acc[2]);
    acc[3] = WMMA_F32(a, b3, acc[3]);
  }

#pragma unroll
  for (int n = 0; n < 4; ++n) {
    const int   colI = n * 16 + m;
    const float bv   = bias[colI];
#pragma unroll
    for (int rr = 0; rr < 8; ++rr) {
      const int nrow = tn + rr + half * 8;
      float v = acc[n][rr] + bv;
      v = v > 0.0f ? v : 0.0f;
      if (nrow < N)
        out[(size_t)nrow * HID_NF + colI] = v;
    }
  }
}

extern "C" void kernel_launch(void* const* d_in, const int* in_sizes, int n_in,
                              void* d_out, int out_size, void* d_ws, size_t ws_size,
                              hipStream_t stream) {
  const float* h      = (const float*)d_in[0];
  const int*   eidx   = (const int*)  d_in[1];
  const float* edge_w = (const float*)d_in[2];
  const float* edge_b = (const float*)d_in[3];
  const float* node_w = (const float*)d_in[4];
  const float* node_b = (const float*)d_in[5];
  float* out = (float*)d_out;

  const int N = in_sizes[0] / IN_NF;
  const int E = in_sizes[1] / 2;
  float* agg = (float*)d_ws;                       // [N, HID_NF] f32 scratch

  const int n4 = (N * HID_NF) / 4;
  gnn_zero_kernel<<<1024, 256, 0, stream>>>(agg, n4);

  const int eblocks = (E + 127) / 128;             // 128 edges per block (8 waves)
  gnn_edge_kernel<<<eblocks, 256, 0, stream>>>(h, eidx, edge_w, edge_b, agg, E);

  const int nblocks = (N + 127) / 128;
  gnn_node_kernel<<<nblocks, 256, 0, stream>>>(h, agg, node_w, node_b, out, N);
}